// MobileMQA_76390288326727
// MI455X (gfx1250) — compile-verified
//
#include <hip/hip_runtime.h>
#include <hip/hip_bf16.h>
#include <stdint.h>

#define C_CH   256
#define NHEAD  8
#define DH     32
#define NPIX   1024   // 32*32
#define MPIX   256    // 16*16
#define NGRP   32
#define EPSV   1e-5f

typedef __attribute__((ext_vector_type(2)))  float v2f;
typedef __attribute__((ext_vector_type(8)))  float v8f;
typedef __attribute__((ext_vector_type(4)))  unsigned int u32x4;
typedef __attribute__((ext_vector_type(8)))  int i32x8;
typedef __attribute__((ext_vector_type(4)))  int i32x4;

// -------- fp32 WMMA wrapper (V_WMMA_F32_16X16X4_F32) --------
__device__ __forceinline__ v8f wmma_f32_4(v2f a, v2f b, v8f c) {
  return __builtin_amdgcn_wmma_f32_16x16x4_f32(false, a, false, b, (short)0, c,
                                               false, false);
}

// -------- TDM 2D tile load: global -> LDS (D# per ISA ch.8) --------
__device__ __forceinline__ void tdm_load_2d(unsigned lds_addr, const void* gptr,
                                            unsigned dim0, unsigned dim1,
                                            unsigned tile0, unsigned tile1,
                                            unsigned long long stride0_elems) {
#if __has_builtin(__builtin_amdgcn_tensor_load_to_lds)
  unsigned long long ga = (unsigned long long)(uintptr_t)gptr;
  u32x4 g0;
  g0[0] = 1u;                                   // count=1, user descriptor
  g0[1] = lds_addr;                             // LDS byte address
  g0[2] = (unsigned)(ga & 0xFFFFFFFFull);       // global_addr[31:0]
  g0[3] = (unsigned)((ga >> 32) & 0x1FFFFFFull) // global_addr[56:32]
          | (2u << 30);                         // type=2 ("image")
  i32x8 g1;
  g1[0] = (int)(2u << 16);                                  // data_size=4B
  g1[1] = (int)((dim0 & 0xFFFFu) << 16);                    // tensor_dim0[15:0]
  g1[2] = (int)(((dim0 >> 16) & 0xFFFFu) | ((dim1 & 0xFFFFu) << 16));
  g1[3] = (int)(((dim1 >> 16) & 0xFFFFu) | ((tile0 & 0xFFFFu) << 16));
  g1[4] = (int)(tile1 & 0xFFFFu);                           // tile_dim1, tile_dim2=0
  g1[5] = (int)(stride0_elems & 0xFFFFFFFFull);             // dim0 stride lo
  g1[6] = (int)((stride0_elems >> 32) & 0xFFFFull);         // dim0 stride hi
  g1[7] = 0;
  i32x4 z4 = {0, 0, 0, 0};
#if defined(__clang_major__) && (__clang_major__ >= 23)
  i32x8 z8 = {0, 0, 0, 0, 0, 0, 0, 0};
  __builtin_amdgcn_tensor_load_to_lds(g0, g1, z4, z4, z8, 0);
#else
  __builtin_amdgcn_tensor_load_to_lds(g0, g1, z4, z4, 0);
#endif
#endif
}

// ============================================================
// Kernel 1: fused q/k/v conv1x1 as GEMM  Y[256,1024] = W[256,256] * X[256,1024]
// grid: B * 3 * (C/16) * (N/128), block 256 (8 waves), wave -> one 16x16 tile
// ============================================================
__global__ __launch_bounds__(256) void qkv_gemm_kernel(
    const float* __restrict__ x,
    const float* __restrict__ wq, const float* __restrict__ bq,
    const float* __restrict__ wk, const float* __restrict__ bk,
    const float* __restrict__ wv, const float* __restrict__ bv,
    float* __restrict__ qo, float* __restrict__ ko, float* __restrict__ vo) {
  int bid = blockIdx.x;
  int nt = bid & 7;  bid >>= 3;
  int ot = bid & 15; bid >>= 4;
  int which = bid % 3;
  int b = bid / 3;

  const float* W    = (which == 0) ? wq : (which == 1) ? wk : wv;
  const float* bias = (which == 0) ? bq : (which == 1) ? bk : bv;
  float* out        = (which == 0) ? qo : (which == 1) ? ko : vo;

  const float* X = x + (size_t)b * C_CH * NPIX;
  out += (size_t)b * C_CH * NPIX;

  int lane = threadIdx.x & 31;
  int wave = threadIdx.x >> 5;
  int col  = lane & 15;
  int half = lane >> 4;
  int o0 = ot * 16;
  int n0 = nt * 128 + wave * 16;

  const float* Wr = W + (size_t)(o0 + col) * C_CH + 2 * half;

  v8f acc = {0.f, 0.f, 0.f, 0.f, 0.f, 0.f, 0.f, 0.f};
  for (int k0 = 0; k0 < C_CH; k0 += 4) {
    if (k0 + 4 < C_CH)
      __builtin_prefetch(&X[(size_t)(k0 + 4 + half) * NPIX + n0 + col], 0, 1);
    v2f a, bb;
    a[0]  = Wr[k0 + 0];                                 // A[row=col][k=2h+0]
    a[1]  = Wr[k0 + 1];                                 // A[row=col][k=2h+1]
    bb[0] = X[(size_t)(k0 + 0 + half) * NPIX + n0 + col];  // B[k=2*0+h][n=col]
    bb[1] = X[(size_t)(k0 + 2 + half) * NPIX + n0 + col];  // B[k=2*1+h][n=col]
    acc = wmma_f32_4(a, bb, acc);
  }
#pragma unroll
  for (int r = 0; r < 8; ++r) {
    int o = o0 + r + 8 * half;                          // C tile: row o, col n
    out[(size_t)o * NPIX + n0 + col] = acc[r] + bias[o];
  }
}

// ============================================================
// Kernel 2: depthwise 3x3 stride-2 + GroupNorm(8ch) + SiLU
// grid: B * NGRP, block 256 (each thread: 8 channels x 1 pixel)
// ============================================================
__global__ __launch_bounds__(256) void dw_gn1_kernel(
    const float* __restrict__ inp,   // (B,C,32,32)
    const float* __restrict__ dw,    // (C,9)
    const float* __restrict__ gs, const float* __restrict__ gb,
    float* __restrict__ outp) {      // (B,C,16,16)
  __shared__ float redS[256];
  __shared__ float redQ[256];
  __shared__ float s_mu, s_rstd;

  int g = blockIdx.x & 31;
  int b = blockIdx.x >> 5;
  int tid = threadIdx.x;
  int ho = tid >> 4, wo = tid & 15;

  float vals[8];
  float s = 0.f, s2 = 0.f;
#pragma unroll
  for (int t = 0; t < 8; ++t) {
    int c = g * 8 + t;
    const float* ib = inp + ((size_t)b * C_CH + c) * NPIX;
    const float* wgt = dw + c * 9;
    float acc = 0.f;
#pragma unroll
    for (int kh = 0; kh < 3; ++kh) {
      int h = 2 * ho - 1 + kh;
      if (h < 0 || h > 31) continue;
#pragma unroll
      for (int kw = 0; kw < 3; ++kw) {
        int w = 2 * wo - 1 + kw;
        if (w < 0 || w > 31) continue;
        acc += ib[h * 32 + w] * wgt[kh * 3 + kw];
      }
    }
    vals[t] = acc;
    s += acc;
    s2 += acc * acc;
  }
  redS[tid] = s; redQ[tid] = s2;
  __syncthreads();
  for (int off = 128; off > 0; off >>= 1) {
    if (tid < off) { redS[tid] += redS[tid + off]; redQ[tid] += redQ[tid + off]; }
    __syncthreads();
  }
  if (tid == 0) {
    float mu  = redS[0] * (1.f / 2048.f);
    float var = redQ[0] * (1.f / 2048.f) - mu * mu;
    s_mu = mu;
    s_rstd = rsqrtf(var + EPSV);
  }
  __syncthreads();
  float mu = s_mu, rstd = s_rstd;
#pragma unroll
  for (int t = 0; t < 8; ++t) {
    int c = g * 8 + t;
    float y = (vals[t] - mu) * rstd * gs[c] + gb[c];
    y = y * (1.f / (1.f + __expf(-y)));   // SiLU
    outp[((size_t)b * C_CH + c) * MPIX + tid] = y;
  }
}

// ============================================================
// Kernel 3: pointwise conv GEMM  Y[256,256] = PW[256,256] * Y1[256,256]
// grid: B * (C/16) * (M/128), block 256
// ============================================================
__global__ __launch_bounds__(256) void pw_gemm_kernel(
    const float* __restrict__ y1, const float* __restrict__ W,
    float* __restrict__ y2) {
  int bid = blockIdx.x;
  int nt = bid & 1;  bid >>= 1;
  int ot = bid & 15; bid >>= 4;
  int b = bid;

  const float* X = y1 + (size_t)b * C_CH * MPIX;
  float* out     = y2 + (size_t)b * C_CH * MPIX;

  int lane = threadIdx.x & 31;
  int wave = threadIdx.x >> 5;
  int col  = lane & 15;
  int half = lane >> 4;
  int o0 = ot * 16;
  int n0 = nt * 128 + wave * 16;

  const float* Wr = W + (size_t)(o0 + col) * C_CH + 2 * half;

  v8f acc = {0.f, 0.f, 0.f, 0.f, 0.f, 0.f, 0.f, 0.f};
  for (int k0 = 0; k0 < C_CH; k0 += 4) {
    v2f a, bb;
    a[0]  = Wr[k0 + 0];
    a[1]  = Wr[k0 + 1];
    bb[0] = X[(size_t)(k0 + 0 + half) * MPIX + n0 + col];
    bb[1] = X[(size_t)(k0 + 2 + half) * MPIX + n0 + col];
    acc = wmma_f32_4(a, bb, acc);
  }
#pragma unroll
  for (int r = 0; r < 8; ++r) {
    int o = o0 + r + 8 * half;
    out[(size_t)o * MPIX + n0 + col] = acc[r];
  }
}

// ============================================================
// Kernel 4: GroupNorm(8ch over 16x16) + SiLU on GEMM output
// ============================================================
__global__ __launch_bounds__(256) void gn2_kernel(
    const float* __restrict__ inp, const float* __restrict__ gs,
    const float* __restrict__ gb, float* __restrict__ outp) {
  __shared__ float redS[256];
  __shared__ float redQ[256];
  __shared__ float s_mu, s_rstd;

  int g = blockIdx.x & 31;
  int b = blockIdx.x >> 5;
  int tid = threadIdx.x;

  float vals[8];
  float s = 0.f, s2 = 0.f;
#pragma unroll
  for (int t = 0; t < 8; ++t) {
    int c = g * 8 + t;
    float v = inp[((size_t)b * C_CH + c) * MPIX + tid];
    vals[t] = v;
    s += v;
    s2 += v * v;
  }
  redS[tid] = s; redQ[tid] = s2;
  __syncthreads();
  for (int off = 128; off > 0; off >>= 1) {
    if (tid < off) { redS[tid] += redS[tid + off]; redQ[tid] += redQ[tid + off]; }
    __syncthreads();
  }
  if (tid == 0) {
    float mu  = redS[0] * (1.f / 2048.f);
    float var = redQ[0] * (1.f / 2048.f) - mu * mu;
    s_mu = mu;
    s_rstd = rsqrtf(var + EPSV);
  }
  __syncthreads();
  float mu = s_mu, rstd = s_rstd;
#pragma unroll
  for (int t = 0; t < 8; ++t) {
    int c = g * 8 + t;
    float y = (vals[t] - mu) * rstd * gs[c] + gb[c];
    y = y * (1.f / (1.f + __expf(-y)));
    outp[((size_t)b * C_CH + c) * MPIX + tid] = y;
  }
}

// ============================================================
// Kernel 5: attention per (b, head). K_h/V_h (32x256 fp32) staged into LDS
// via TDM; WMMA f32 for QK^T and PV; wave32 shfl softmax.
// block = 128 (4 waves); each wave owns 16 q-rows at a time.
// ============================================================
__global__ __launch_bounds__(128) void attn_kernel(
    const float* __restrict__ q, const float* __restrict__ kd,
    const float* __restrict__ vd, float* __restrict__ out) {
  __shared__ float lds_k[DH * MPIX];        // [d][m]  32 KB
  __shared__ float lds_v[DH * MPIX];        // [d][m]  32 KB
  __shared__ float lds_p[4][16 * MPIX];     // per-wave P slab, 64 KB

  int b = blockIdx.x >> 3;
  int h = blockIdx.x & 7;
  const float* qb = q  + ((size_t)b * C_CH + h * DH) * NPIX;
  const float* kb = kd + ((size_t)b * C_CH + h * DH) * MPIX;
  const float* vb = vd + ((size_t)b * C_CH + h * DH) * MPIX;
  float* ob       = out + ((size_t)b * C_CH + h * DH) * NPIX;

#if __has_builtin(__builtin_amdgcn_tensor_load_to_lds)
  // Only wave 0 issues the two TDM descriptors (scalarized condition).
  if (__builtin_amdgcn_readfirstlane(threadIdx.x >> 5) == 0) {
    tdm_load_2d((unsigned)(uintptr_t)&lds_k[0], kb, MPIX, DH, MPIX, DH, MPIX);
    tdm_load_2d((unsigned)(uintptr_t)&lds_v[0], vb, MPIX, DH, MPIX, DH, MPIX);
    __builtin_amdgcn_s_wait_tensorcnt(0);
  }
#else
  for (int i = threadIdx.x; i < DH * MPIX; i += blockDim.x) {
    lds_k[i] = kb[i];
    lds_v[i] = vb[i];
  }
#endif
  __syncthreads();

  int lane = threadIdx.x & 31;
  int wave = threadIdx.x >> 5;
  int col  = lane & 15;
  int half = lane >> 4;
  const float scale = 0.17677669529663687f;   // 1/sqrt(32)

  float* P = &lds_p[wave][0];

  for (int nb = wave; nb < 64; nb += 4) {
    int n0 = nb * 16;

    // ---- scores: S[16 x 256] = Q_tile(16x32) * K^T ----
    v8f sc[16];
#pragma unroll
    for (int mt = 0; mt < 16; ++mt) {
      v8f acc = {0.f, 0.f, 0.f, 0.f, 0.f, 0.f, 0.f, 0.f};
#pragma unroll
      for (int d0 = 0; d0 < DH; d0 += 4) {
        v2f a, bb;
        a[0]  = qb[(size_t)(d0 + 2 * half + 0) * NPIX + n0 + col];
        a[1]  = qb[(size_t)(d0 + 2 * half + 1) * NPIX + n0 + col];
        bb[0] = lds_k[(d0 + 0 + half) * MPIX + mt * 16 + col];
        bb[1] = lds_k[(d0 + 2 + half) * MPIX + mt * 16 + col];
        acc = wmma_f32_4(a, bb, acc);
      }
      sc[mt] = acc;
    }

    // ---- softmax over m (row = (r, half); spread over 16 lanes x 16 tiles) ----
#pragma unroll
    for (int r = 0; r < 8; ++r) {
      float mx = -3.0e38f;
#pragma unroll
      for (int mt = 0; mt < 16; ++mt) {
        sc[mt][r] *= scale;
        mx = fmaxf(mx, sc[mt][r]);
      }
      for (int off = 1; off < 16; off <<= 1)
        mx = fmaxf(mx, __shfl_xor(mx, off, 32));
      float sum = 0.f;
#pragma unroll
      for (int mt = 0; mt < 16; ++mt) {
        float e = __expf(sc[mt][r] - mx);
        sc[mt][r] = e;
        sum += e;
      }
      for (int off = 1; off < 16; off <<= 1)
        sum += __shfl_xor(sum, off, 32);
      float inv = 1.f / sum;
#pragma unroll
      for (int mt = 0; mt < 16; ++mt) sc[mt][r] *= inv;
    }

    // ---- stage P in LDS row-major [n'][m] (C-layout -> A-layout bridge) ----
#pragma unroll
    for (int mt = 0; mt < 16; ++mt)
#pragma unroll
      for (int r = 0; r < 8; ++r)
        P[(r + 8 * half) * MPIX + mt * 16 + col] = sc[mt][r];

    // ---- out: O[16 x 32] = P(16x256) * V(256x32) ----
#pragma unroll
    for (int dt = 0; dt < 2; ++dt) {
      v8f acc = {0.f, 0.f, 0.f, 0.f, 0.f, 0.f, 0.f, 0.f};
      for (int m0 = 0; m0 < MPIX; m0 += 4) {
        v2f a, bb;
        a[0]  = P[col * MPIX + m0 + 2 * half + 0];
        a[1]  = P[col * MPIX + m0 + 2 * half + 1];
        bb[0] = lds_v[(dt * 16 + col) * MPIX + m0 + 0 + half];
        bb[1] = lds_v[(dt * 16 + col) * MPIX + m0 + 2 + half];
        acc = wmma_f32_4(a, bb, acc);
      }
#pragma unroll
      for (int r = 0; r < 8; ++r)
        ob[(size_t)(dt * 16 + col) * NPIX + n0 + r + 8 * half] = acc[r];
    }
  }
}

// ============================================================
extern "C" void kernel_launch(void* const* d_in, const int* in_sizes, int n_in,
                              void* d_out, int out_size, void* d_ws,
                              size_t ws_size, hipStream_t stream) {
  const float* x      = (const float*)d_in[0];
  const float* wq     = (const float*)d_in[1];
  const float* bq     = (const float*)d_in[2];
  const float* wk     = (const float*)d_in[3];
  const float* bk     = (const float*)d_in[4];
  const float* wv     = (const float*)d_in[5];
  const float* bv     = (const float*)d_in[6];
  const float* k_dw   = (const float*)d_in[7];
  const float* k_gn1s = (const float*)d_in[8];
  const float* k_gn1b = (const float*)d_in[9];
  const float* k_pw   = (const float*)d_in[10];
  const float* k_gn2s = (const float*)d_in[11];
  const float* k_gn2b = (const float*)d_in[12];
  const float* v_dw   = (const float*)d_in[13];
  const float* v_gn1s = (const float*)d_in[14];
  const float* v_gn1b = (const float*)d_in[15];
  const float* v_pw   = (const float*)d_in[16];
  const float* v_gn2s = (const float*)d_in[17];
  const float* v_gn2b = (const float*)d_in[18];
  float* out = (float*)d_out;
  float* ws  = (float*)d_ws;

  const size_t SZ_FULL = (size_t)32 * C_CH * NPIX;  // 8 Mi floats
  const size_t SZ_DS   = (size_t)32 * C_CH * MPIX;  // 2 Mi floats
  float* qw  = ws;
  float* kf  = qw + SZ_FULL;
  float* vf  = kf + SZ_FULL;
  float* y1k = vf + SZ_FULL;
  float* y1v = y1k + SZ_DS;
  float* y2k = y1v + SZ_DS;
  float* y2v = y2k + SZ_DS;
  float* kdp = y2v + SZ_DS;
  float* vdp = kdp + SZ_DS;

  qkv_gemm_kernel<<<32 * 3 * 16 * 8, 256, 0, stream>>>(
      x, wq, bq, wk, bk, wv, bv, qw, kf, vf);

  dw_gn1_kernel<<<32 * NGRP, 256, 0, stream>>>(kf, k_dw, k_gn1s, k_gn1b, y1k);
  dw_gn1_kernel<<<32 * NGRP, 256, 0, stream>>>(vf, v_dw, v_gn1s, v_gn1b, y1v);

  pw_gemm_kernel<<<32 * 16 * 2, 256, 0, stream>>>(y1k, k_pw, y2k);
  pw_gemm_kernel<<<32 * 16 * 2, 256, 0, stream>>>(y1v, v_pw, y2v);

  gn2_kernel<<<32 * NGRP, 256, 0, stream>>>(y2k, k_gn2s, k_gn2b, kdp);
  gn2_kernel<<<32 * NGRP, 256, 0, stream>>>(y2v, v_gn2s, v_gn2b, vdp);

  attn_kernel<<<32 * NHEAD, 128, 0, stream>>>(qw, kdp, vdp, out);
}